// SepKANLayer1D_64192581206112
// MI455X (gfx1250) — compile-verified
//
#include <hip/hip_runtime.h>
#include <cstdint>
#include <cstddef>

// ---------------------------------------------------------------------------
// SepKANLayer1D for MI455X (gfx1250).
// Memory-bound kernel (~0.6 FLOP/byte, ~688 MB total => ~30us floor @23.3TB/s).
// The 512 per-token spline-coef rows are streamed through LDS with the CDNA5
// Tensor Data Mover (tensor_load_to_lds), double-buffered, one DMA per wave
// per chunk (TENSORcnt is per-wave). uw/rw/x/out use non-temporal direct
// VMEM (each byte touched exactly once). WMMA is mathematically inapplicable
// (per-token weights; output quadratic in w).
// ---------------------------------------------------------------------------

#define IN_C_    8
#define OUT_C_   8
#define NB_      8
#define SIZE_    640           // 8*8*8 coef + 64 uw + 64 rw
#define NTOK_    65536         // tokens per batch
#define NBATCH_  4
#define TPB_     128           // threads (= tokens) per block -> 4 wave32
#define CHROWS_  64            // w-rows per chunk == one input channel i
#define WROWS_   16            // rows DMA'd by each of the 4 waves
#define NCH_     8             // coef chunks (512 rows / 64)

typedef uint32_t u32x4 __attribute__((ext_vector_type(4)));
typedef uint32_t u32x8 __attribute__((ext_vector_type(8)));

// ---- cubic B-spline, uniform knots t_j = 0.2*(j-3), j=0..11 ----------------
// Knots and reciprocal denominators are compile-time constants: folds to
// compares + cndmask + FMA chains (no runtime divides).
__device__ __forceinline__ void bspline8(float x, float B[NB_]) {
  float kn[12];
#pragma unroll
  for (int j = 0; j < 12; ++j) kn[j] = 0.2f * (float)(j - 3);
  float b[11];
#pragma unroll
  for (int j = 0; j < 11; ++j) b[j] = (x >= kn[j] && x < kn[j + 1]) ? 1.0f : 0.0f;
#pragma unroll
  for (int p = 1; p <= 3; ++p) {
#pragma unroll
    for (int j = 0; j < 11 - p; ++j) {
      const float l = (x - kn[j])         * (1.0f / (kn[j + p]     - kn[j]));
      const float r = (kn[j + p + 1] - x) * (1.0f / (kn[j + p + 1] - kn[j + 1]));
      b[j] = fmaf(l, b[j], r * b[j + 1]);
    }
  }
#pragma unroll
  for (int j = 0; j < NB_; ++j) B[j] = b[j];
}

// ---- TDM: DMA a [WROWS_ x TPB_] f32 tile (row stride NTOK_) into LDS -------
// One DMA per wave per chunk. D# per CDNA5 ISA:
//   group0: count=1 | lds_addr | global_addr(57b) | type=2
//   group1: data_size=4B, tensor_dim0=65536, tensor_dim1=16,
//           tile_dim0=128, tile_dim1=16, tensor_dim0_stride=65536
__device__ __forceinline__ void tdm_issue_rows(const float* gsrc, const float* ldsdst) {
  const uint64_t ga = (uint64_t)(uintptr_t)gsrc;
  const uint32_t la = (uint32_t)(uintptr_t)ldsdst;   // low 32 bits = LDS byte offset
  const u32x4 g0 = { 1u,                                   // count=1 (valid user D#)
                     la,                                   // lds_addr
                     (uint32_t)ga,                         // global_addr[31:0]
                     ((uint32_t)(ga >> 32) & 0x01FFFFFFu)  // global_addr[56:32]
                       | (2u << 30) };                     // type=2 ("image")
  const u32x8 g1 = { 0x00020000u,                   // wg_mask=0, data_size=2 (4B)
                     0u,                             // atomic_barrier=0, dim0[15:0]=0
                     ((uint32_t)WROWS_ << 16) | 1u,  // dim0[31:16]=1 (65536), dim1=16
                     ((uint32_t)TPB_) << 16,         // dim1 hi=0, tile_dim0=128
                     (uint32_t)WROWS_,               // tile_dim1=16, tile_dim2=0
                     (uint32_t)NTOK_,                // tensor_dim0_stride[31:0]=65536
                     0u,                             // stride0 hi=0, stride1 lo=0
                     0u };                           // stride1 hi=0
  asm volatile("tensor_load_to_lds %0, %1" :: "s"(g0), "s"(g1) : "memory");
}

__global__ __launch_bounds__(TPB_) void kan1d_kernel(const float* __restrict__ x,
                                                     const float* __restrict__ w,
                                                     float* __restrict__ out) {
  __shared__ float buf[2][CHROWS_ * TPB_];   // 2 x 32 KB double buffer

  const int tid  = (int)threadIdx.x;
  const int wave = tid >> 5;                 // 0..3, wave-uniform
  const int b    = (int)blockIdx.y;
  const int n0   = (int)blockIdx.x * TPB_;

  const float* wbase = w + ((size_t)b * SIZE_) * NTOK_ + n0;        // w[b][row][n0..]
  const float* wwave = wbase + (size_t)wave * WROWS_ * NTOK_;       // this wave's rows
  const float* xbase = x + ((size_t)b * IN_C_) * NTOK_ + n0 + tid;
  float*       obase = out + ((size_t)b * OUT_C_) * NTOK_ + n0 + tid;

  // prologue: every wave DMAs its 16-row sub-tile of chunks 0 and 1
  tdm_issue_rows(wwave + (size_t)0 * CHROWS_ * NTOK_, &buf[0][wave * WROWS_ * TPB_]);
  tdm_issue_rows(wwave + (size_t)1 * CHROWS_ * NTOK_, &buf[1][wave * WROWS_ * TPB_]);

  float acc[OUT_C_];
#pragma unroll
  for (int o = 0; o < OUT_C_; ++o) acc[o] = 0.0f;

  for (int i = 0; i < NCH_; ++i) {          // chunk i == input channel i
    if (i < NCH_ - 1) __builtin_amdgcn_s_wait_tensorcnt(1);  // this wave's oldest DMA done
    else              __builtin_amdgcn_s_wait_tensorcnt(0);  // last DMA done
    __syncthreads();                        // all 4 sub-tiles of chunk i visible

    const float* cur = &buf[i & 1][0];

    const float xi = __builtin_nontemporal_load(&xbase[(size_t)i * NTOK_]);
    float Bv[NB_];
    bspline8(xi, Bv);
    const float sil = xi / (1.0f + __expf(-xi));   // silu(x_i)

#pragma unroll
    for (int o = 0; o < OUT_C_; ++o) {
      float sp = 0.0f;
#pragma unroll
      for (int k = 0; k < NB_; ++k)          // coef row s = i*64 + o*8 + k
        sp = fmaf(Bv[k], cur[(o * NB_ + k) * TPB_ + tid], sp);
      const float uwv =
          __builtin_nontemporal_load(&wbase[(size_t)(512 + i * 8 + o) * NTOK_ + tid]);
      const float rwv =
          __builtin_nontemporal_load(&wbase[(size_t)(576 + i * 8 + o) * NTOK_ + tid]);
      acc[o] = fmaf(uwv, sp, fmaf(sil, rwv, acc[o]));
    }

    __syncthreads();                        // everyone done reading buf[i&1]
    if ((i + 2) < NCH_)
      tdm_issue_rows(wwave + (size_t)(i + 2) * CHROWS_ * NTOK_,
                     &buf[i & 1][wave * WROWS_ * TPB_]);
  }

#pragma unroll
  for (int o = 0; o < OUT_C_; ++o)
    __builtin_nontemporal_store(acc[o], &obase[(size_t)o * NTOK_]);
}

extern "C" void kernel_launch(void* const* d_in, const int* in_sizes, int n_in,
                              void* d_out, int out_size, void* d_ws, size_t ws_size,
                              hipStream_t stream) {
  (void)in_sizes; (void)n_in; (void)out_size; (void)d_ws; (void)ws_size;
  const float* x = (const float*)d_in[0];   // (4, 8, 65536) f32
  const float* w = (const float*)d_in[1];   // (4, 640, 65536) f32
  float* out = (float*)d_out;               // (4, 8, 65536) f32
  dim3 grid(NTOK_ / TPB_, NBATCH_);
  kan1d_kernel<<<grid, TPB_, 0, stream>>>(x, w, out);
}